// ValScores_17016660426754
// MI455X (gfx1250) — compile-verified
//
#include <hip/hip_runtime.h>
#include <hip/hip_bf16.h>

// ---------------------------------------------------------------------------
// ValScores update:  per-class mean of preds (segment reduction by label)
// followed by EMA with val_preds.
//
// Strategy (MI455X, gfx1250, wave32):
//  * Bandwidth-bound: preds = 524 MB, streamed exactly once (~22.5us floor at
//    23.3 TB/s). No global float atomics in the hot path.
//  * Counting-sort (row,label) pairs by label (labels are only ~1 MB traffic).
//  * Segment sum as a one-hot matmul on V_WMMA_F32_16X16X4_F32 in full f32:
//    each block owns a [16-class x 64-col] output tile and contracts only the
//    contiguous sorted rows of its 16 classes. A-tile (16x4 one-hot) is built
//    in registers from sorted labels; B-tile loads are coalesced GVS-mode
//    32-bit scaled offsets. Ragged K edge handled in a branch-free,
//    wave-uniform tail so EXEC is all-ones at every WMMA.
//  * 4 waves split-K per block, LDS reduction, EMA finalize fused.
// ---------------------------------------------------------------------------

#define NUM_CLASSES 1000
#define CLASS_TILES 63              // 63 * 16 = 1008 >= 1000
#define PAD_CLASSES (CLASS_TILES * 16)
#define COL_STRIPS  16              // 16 * 64 = 1024 >= 1000
#define GAMMA 0.9f

typedef float v2f __attribute__((ext_vector_type(2)));
typedef float v8f __attribute__((ext_vector_type(8)));

// workspace layout (bytes)
#define WS_COUNTS   0                       // PAD_CLASSES ints
#define WS_OFFSETS  4096                    // PAD_CLASSES+1 ints
#define WS_CURSOR   8192                    // PAD_CLASSES ints
#define WS_SORTED   12288                   // N int2 (row idx, label)

__device__ __forceinline__ int imin(int a, int b) { return a < b ? a : b; }
__device__ __forceinline__ int imax(int a, int b) { return a > b ? a : b; }

// --- kernel 1: zero counters --------------------------------------------
__global__ void k_zero(int* counts, int* cursor) {
    int i = blockIdx.x * blockDim.x + threadIdx.x;
    if (i < PAD_CLASSES) { counts[i] = 0; cursor[i] = 0; }
}

// --- kernel 2: label histogram (int atomics -> deterministic counts) ----
__global__ void k_hist(const int* __restrict__ labels, int n, int* counts) {
    int i = blockIdx.x * blockDim.x + threadIdx.x;
    if (i < n) atomicAdd(&counts[labels[i]], 1);
}

// --- kernel 3: exclusive scan (tiny: 1000 elements, single thread) ------
__global__ void k_scan(const int* __restrict__ counts, int* offsets) {
    if (threadIdx.x == 0 && blockIdx.x == 0) {
        int s = 0;
        for (int c = 0; c < NUM_CLASSES; ++c) { offsets[c] = s; s += counts[c]; }
        for (int c = NUM_CLASSES; c <= PAD_CLASSES; ++c) offsets[c] = s;
    }
}

// --- kernel 4: scatter (row, label) pairs into label-sorted order -------
// (intra-class order is atomic-arrival order; per-class membership is exact)
__global__ void k_scatter(const int* __restrict__ labels, int n,
                          const int* __restrict__ offsets, int* cursor,
                          int2* __restrict__ sortedPair) {
    int i = blockIdx.x * blockDim.x + threadIdx.x;
    if (i < n) {
        int lab = labels[i];
        int p = offsets[lab] + atomicAdd(&cursor[lab], 1);
        sortedPair[p] = make_int2(i, lab);      // one b64 store
    }
}

// --- kernel 5: WMMA segment-sum + fused EMA finalize --------------------
// grid = (CLASS_TILES, COL_STRIPS), block = 128 threads (4 waves).
// Wave w contracts rows k = start + 4*w + 16*i (split-K), accumulating a
// [16 x 64] f32 tile via 4 accumulators of v_wmma_f32_16x16x4_f32.
__global__ __launch_bounds__(128)
void k_segsum_wmma_ema(const float* __restrict__ preds,
                       const float* __restrict__ valp,
                       const int* __restrict__ counts,
                       const int* __restrict__ offsets,
                       const int2* __restrict__ sortedPair,
                       float* __restrict__ out) {
    __shared__ float partial[4][16][64];   // 16 KB

    const int lane = threadIdx.x & 31;
    const int wave = threadIdx.x >> 5;
    const int half = lane >> 4;            // 0: K rows {0,1}; 1: K rows {2,3}
    const int n16  = lane & 15;            // A: M index, B/D: N index
    const int classBase = blockIdx.x * 16;
    const int colBase   = blockIdx.y * 64;

    const int start = offsets[classBase];
    const int end   = offsets[classBase + 16];
    const int myClass = classBase + n16;

    // loop-invariant clamped column offsets (32-bit -> GVS scale_offset loads)
    const int coff0 = imin(colBase + 0 * 16 + n16, NUM_CLASSES - 1);
    const int coff1 = imin(colBase + 1 * 16 + n16, NUM_CLASSES - 1);
    const int coff2 = imin(colBase + 2 * 16 + n16, NUM_CLASSES - 1);
    const int coff3 = imin(colBase + 3 * 16 + n16, NUM_CLASSES - 1);

    v8f acc0 = {}, acc1 = {}, acc2 = {}, acc3 = {};

    int k0 = start + wave * 4;
    int kk = k0 + half * 2;

    // ---- main loop: full K-chunks, no clamps, no divergence -------------
#pragma unroll 2
    for (; k0 + 4 <= end; k0 += 16, kk += 16) {
        const int2 p0 = sortedPair[kk];          // coalesced b64
        const int2 p1 = sortedPair[kk + 1];
        __builtin_prefetch(&sortedPair[imin(kk + 64, end - 1)], 0, 3);

        v2f A;                                   // one-hot A[m,k]
        A.x = (p0.y == myClass) ? 1.0f : 0.0f;
        A.y = (p1.y == myClass) ? 1.0f : 0.0f;

        const int b0 = p0.x * NUM_CLASSES;       // 32-bit element offsets
        const int b1 = p1.x * NUM_CLASSES;

        v2f B;
        B.x = preds[b0 + coff0]; B.y = preds[b1 + coff0];
        acc0 = __builtin_amdgcn_wmma_f32_16x16x4_f32(false, A, false, B,
                                                     (short)0, acc0, false, false);
        B.x = preds[b0 + coff1]; B.y = preds[b1 + coff1];
        acc1 = __builtin_amdgcn_wmma_f32_16x16x4_f32(false, A, false, B,
                                                     (short)0, acc1, false, false);
        B.x = preds[b0 + coff2]; B.y = preds[b1 + coff2];
        acc2 = __builtin_amdgcn_wmma_f32_16x16x4_f32(false, A, false, B,
                                                     (short)0, acc2, false, false);
        B.x = preds[b0 + coff3]; B.y = preds[b1 + coff3];
        acc3 = __builtin_amdgcn_wmma_f32_16x16x4_f32(false, A, false, B,
                                                     (short)0, acc3, false, false);
    }

    // ---- ragged tail: wave-uniform branch, branch-free lane guards ------
    if (k0 < end) {
        const int kc0 = imin(kk,     end - 1);
        const int kc1 = imin(kk + 1, end - 1);
        const int2 p0 = sortedPair[kc0];
        const int2 p1 = sortedPair[kc1];
        const int l0 = (kk     < end) ? p0.y : -1;   // v_cndmask, no branch
        const int l1 = (kk + 1 < end) ? p1.y : -1;

        v2f A;
        A.x = (l0 == myClass) ? 1.0f : 0.0f;
        A.y = (l1 == myClass) ? 1.0f : 0.0f;

        const int b0 = p0.x * NUM_CLASSES;
        const int b1 = p1.x * NUM_CLASSES;

        v2f B;
        B.x = preds[b0 + coff0]; B.y = preds[b1 + coff0];
        acc0 = __builtin_amdgcn_wmma_f32_16x16x4_f32(false, A, false, B,
                                                     (short)0, acc0, false, false);
        B.x = preds[b0 + coff1]; B.y = preds[b1 + coff1];
        acc1 = __builtin_amdgcn_wmma_f32_16x16x4_f32(false, A, false, B,
                                                     (short)0, acc1, false, false);
        B.x = preds[b0 + coff2]; B.y = preds[b1 + coff2];
        acc2 = __builtin_amdgcn_wmma_f32_16x16x4_f32(false, A, false, B,
                                                     (short)0, acc2, false, false);
        B.x = preds[b0 + coff3]; B.y = preds[b1 + coff3];
        acc3 = __builtin_amdgcn_wmma_f32_16x16x4_f32(false, A, false, B,
                                                     (short)0, acc3, false, false);
    }

    // ---- spill per-wave partial tiles to LDS ----------------------------
    // D layout: VGPR i holds M=i (lanes 0-15) / M=i+8 (lanes 16-31); N=lane%16
#pragma unroll
    for (int i = 0; i < 8; ++i) {
        const int m = i + half * 8;
        partial[wave][m][0 * 16 + n16] = acc0[i];
        partial[wave][m][1 * 16 + n16] = acc1[i];
        partial[wave][m][2 * 16 + n16] = acc2[i];
        partial[wave][m][3 * 16 + n16] = acc3[i];
    }
    __syncthreads();

    // ---- reduce the 4 split-K partials and fuse the EMA finalize --------
#pragma unroll
    for (int e = threadIdx.x; e < 16 * 64; e += 128) {
        const int m  = e >> 6;
        const int nn = e & 63;
        const int c   = classBase + m;
        const int col = colBase + nn;
        if (c < NUM_CLASSES && col < NUM_CLASSES) {
            const float s = partial[0][m][nn] + partial[1][m][nn]
                          + partial[2][m][nn] + partial[3][m][nn];
            const int cnt = counts[c];
            const float vp = valp[c * NUM_CLASSES + col];
            const float mean = s / (float)imax(cnt, 1);
            out[c * NUM_CLASSES + col] =
                (cnt > 0) ? ((1.0f - GAMMA) * mean + GAMMA * vp) : vp;
        }
    }
}

// ---------------------------------------------------------------------------
extern "C" void kernel_launch(void* const* d_in, const int* in_sizes, int n_in,
                              void* d_out, int out_size, void* d_ws, size_t ws_size,
                              hipStream_t stream) {
    (void)n_in; (void)out_size; (void)ws_size;

    const float* preds  = (const float*)d_in[0];
    const int*   labels = (const int*)d_in[1];
    const float* valp   = (const float*)d_in[2];
    float*       out    = (float*)d_out;

    const int n = in_sizes[1];              // number of rows (labels count)

    char* ws = (char*)d_ws;
    int*  counts     = (int*)(ws + WS_COUNTS);
    int*  offsets    = (int*)(ws + WS_OFFSETS);
    int*  cursor     = (int*)(ws + WS_CURSOR);
    int2* sortedPair = (int2*)(ws + WS_SORTED);

    k_zero<<<(PAD_CLASSES + 255) / 256, 256, 0, stream>>>(counts, cursor);
    k_hist<<<(n + 255) / 256, 256, 0, stream>>>(labels, n, counts);
    k_scan<<<1, 1, 0, stream>>>(counts, offsets);
    k_scatter<<<(n + 255) / 256, 256, 0, stream>>>(labels, n, offsets, cursor,
                                                   sortedPair);

    dim3 grid(CLASS_TILES, COL_STRIPS);
    k_segsum_wmma_ema<<<grid, 128, 0, stream>>>(preds, valp, counts, offsets,
                                                sortedPair, out);
}